// CPUQwen3MoeMoEMLPModule_40450001994271
// MI455X (gfx1250) — compile-verified
//
#include <hip/hip_runtime.h>
#include <hip/hip_bf16.h>

// ---- problem constants (from reference) ----
constexpr int HIDDEN   = 1024;
constexpr int INTER    = 1408;
constexpr int NEXPERTS = 8;
constexpr int TOPK     = 2;
constexpr int TOKENS   = 2048;          // BATCH*SEQ

typedef float v2f __attribute__((ext_vector_type(2)));
typedef float v8f __attribute__((ext_vector_type(8)));

// ---------------------------------------------------------------------------
// Router: one wave (32 lanes) per token. Computes 8 logits, softmax, top-2,
// renormalize; records winners and bumps per-expert counts.
// ---------------------------------------------------------------------------
__global__ __launch_bounds__(256) void moe_router(
    const float* __restrict__ x,        // [T, HIDDEN]
    const float* __restrict__ rw,       // [E, HIDDEN]
    int*  __restrict__ topk_idx,        // [T, 2]
    float* __restrict__ topk_w,         // [T, 2]
    int*  __restrict__ counts)          // [E]
{
    const int wave = threadIdx.x >> 5;
    const int lane = threadIdx.x & 31;
    const int t = blockIdx.x * 8 + wave;
    if (t >= TOKENS) return;

    const float* xr = x + (size_t)t * HIDDEN;
    float s[NEXPERTS];
    #pragma unroll
    for (int e = 0; e < NEXPERTS; ++e) s[e] = 0.0f;

    for (int i = lane; i < HIDDEN; i += 32) {
        const float xv = xr[i];
        #pragma unroll
        for (int e = 0; e < NEXPERTS; ++e)
            s[e] += xv * rw[e * HIDDEN + i];
    }
    #pragma unroll
    for (int e = 0; e < NEXPERTS; ++e)
        for (int off = 16; off; off >>= 1)
            s[e] += __shfl_xor(s[e], off, 32);

    if (lane == 0) {
        float m = s[0];
        #pragma unroll
        for (int e = 1; e < NEXPERTS; ++e) m = fmaxf(m, s[e]);
        float p[NEXPERTS];
        #pragma unroll
        for (int e = 0; e < NEXPERTS; ++e) p[e] = __expf(s[e] - m);

        float best = -1.0f, second = -1.0f;
        int bi = 0, si = 0;
        #pragma unroll
        for (int e = 0; e < NEXPERTS; ++e) {
            if (p[e] > best)        { second = best; si = bi; best = p[e]; bi = e; }
            else if (p[e] > second) { second = p[e]; si = e; }
        }
        const float inv = 1.0f / (best + second);
        topk_idx[t * 2 + 0] = bi;  topk_w[t * 2 + 0] = best   * inv;
        topk_idx[t * 2 + 1] = si;  topk_w[t * 2 + 1] = second * inv;
        atomicAdd(&counts[bi], 1);
        atomicAdd(&counts[si], 1);
    }
}

// ---------------------------------------------------------------------------
// Exclusive prefix over 8 counts -> compacted segment offsets; reset cursors.
// ---------------------------------------------------------------------------
__global__ void moe_scan(const int* __restrict__ counts,
                         int* __restrict__ offsets,
                         int* __restrict__ cursors)
{
    if (threadIdx.x == 0) {
        int acc = 0;
        for (int e = 0; e < NEXPERTS; ++e) {
            offsets[e] = acc; acc += counts[e]; cursors[e] = 0;
        }
        offsets[NEXPERTS] = acc;   // == TOKENS*TOPK == 4096
    }
}

// ---------------------------------------------------------------------------
// Dispatch: scatter (token, weight) pairs into per-expert compacted segments.
// ---------------------------------------------------------------------------
__global__ __launch_bounds__(256) void moe_dispatch(
    const int* __restrict__ topk_idx, const float* __restrict__ topk_w,
    const int* __restrict__ offsets,  int* __restrict__ cursors,
    int* __restrict__ tok_idx, float* __restrict__ tok_w)
{
    const int t = blockIdx.x * 256 + threadIdx.x;
    if (t >= TOKENS) return;
    #pragma unroll
    for (int k = 0; k < TOPK; ++k) {
        const int e = topk_idx[t * 2 + k];
        const int pos = atomicAdd(&cursors[e], 1);
        const int slot = offsets[e] + pos;
        tok_idx[slot] = t;
        tok_w[slot]   = topk_w[t * 2 + k];
    }
}

// ---------------------------------------------------------------------------
// Stage 1: H[seg+m, n] = silu(Xg) * Xu for expert e's gathered tokens.
// Block tile: M=16 tokens x N=128 inter cols; K-loop over HIDDEN in 32-chunks,
// software-pipelined: global loads for chunk i+1 in flight during WMMA of
// chunk i; LDS ping-pong, one barrier per chunk. fp32 WMMA 16x16x4.
// ---------------------------------------------------------------------------
__global__ __launch_bounds__(256) void moe_stage1(
    const float* __restrict__ x,        // [T, HIDDEN]
    const float* __restrict__ wg,       // [E, HIDDEN, INTER]
    const float* __restrict__ wu,       // [E, HIDDEN, INTER]
    const int*  __restrict__ offsets,   // [E+1]
    const int*  __restrict__ tok_idx,   // [T*K]
    float* __restrict__ H)              // [T*K, INTER]
{
    const int e   = blockIdx.z;
    const int n0  = blockIdx.y * 128;
    const int seg = offsets[e];
    const int cnt = offsets[e + 1] - seg;
    const int m0  = blockIdx.x * 16;
    if (m0 >= cnt) return;                      // block-uniform exit (EXEC stays full)

    __shared__ float Alds[2][16][36];           // 16x32 (+4 pad), double buffered
    __shared__ float Bg[2][32][132];            // 32x128 (+4 pad)
    __shared__ float Bu[2][32][132];
    __shared__ int   toks[16];

    const int tid   = threadIdx.x;
    const int lane  = tid & 31;
    const int wave  = tid >> 5;
    const int row16 = lane & 15;
    const int kb    = (lane >> 4) << 1;         // 0 or 2 (A/B fragment K phase)

    // per-thread staging coordinates (chunk-invariant)
    const int a_m = tid >> 3;                   // 0..31 (use <16 via tid<128)
    const int a_c = (tid & 7) << 2;             // 0..28
    const int b_c = lane << 2;                  // 0..124; row = wave + 8*i

    if (tid < 16) {
        const int m = m0 + tid;
        toks[tid] = (m < cnt) ? tok_idx[seg + m] : -1;
    }
    __syncthreads();

    const float* __restrict__ wgE = wg + (size_t)e * HIDDEN * INTER;
    const float* __restrict__ wuE = wu + (size_t)e * HIDDEN * INTER;

    float4 ra, rg[4], ru[4];

    auto load_regs = [&](int K0) {
        if (tid < 128) {
            const int t = toks[a_m];
            ra = make_float4(0.f, 0.f, 0.f, 0.f);
            if (t >= 0) ra = *(const float4*)(x + (size_t)t * HIDDEN + K0 + a_c);
        }
        #pragma unroll
        for (int i = 0; i < 4; ++i) {
            const size_t go = (size_t)(K0 + wave + 8 * i) * INTER + n0 + b_c;
            rg[i] = *(const float4*)(wgE + go);
            ru[i] = *(const float4*)(wuE + go);
        }
        if (K0 + 32 < HIDDEN) {                 // warm L0 one more chunk ahead
            __builtin_prefetch(wgE + (size_t)(K0 + 32 + wave) * INTER + n0 + b_c, 0, 0);
            __builtin_prefetch(wuE + (size_t)(K0 + 32 + wave) * INTER + n0 + b_c, 0, 0);
        }
    };
    auto store_regs = [&](int B) {
        if (tid < 128) *(float4*)&Alds[B][a_m][a_c] = ra;
        #pragma unroll
        for (int i = 0; i < 4; ++i) {
            *(float4*)&Bg[B][wave + 8 * i][b_c] = rg[i];
            *(float4*)&Bu[B][wave + 8 * i][b_c] = ru[i];
        }
    };

    v8f accG = {}; v8f accU = {};
    const int ncolw = (wave << 4) + row16;

    auto compute = [&](int B) {
        #pragma unroll
        for (int kk = 0; kk < 32; kk += 4) {
            v2f a, bg, bu;
            a.x  = Alds[B][row16][kk + kb];
            a.y  = Alds[B][row16][kk + kb + 1];
            bg.x = Bg[B][kk + kb][ncolw];
            bg.y = Bg[B][kk + kb + 1][ncolw];
            bu.x = Bu[B][kk + kb][ncolw];
            bu.y = Bu[B][kk + kb + 1][ncolw];
            accG = __builtin_amdgcn_wmma_f32_16x16x4_f32(
                false, a, false, bg, (short)0, accG, false, false);
            accU = __builtin_amdgcn_wmma_f32_16x16x4_f32(
                false, a, false, bu, (short)0, accU, false, false);
        }
    };

    // pipeline prologue
    load_regs(0);
    store_regs(0);
    __syncthreads();
    load_regs(32);                              // chunk 1 in flight
    int buf = 0;
    for (int k0 = 0; k0 < HIDDEN; k0 += 32) {
        compute(buf);                           // overlaps in-flight global loads
        if (k0 + 32 < HIDDEN) {
            store_regs(buf ^ 1);
            if (k0 + 64 < HIDDEN) load_regs(k0 + 64);
        }
        __syncthreads();
        buf ^= 1;
    }

    // h = silu(g) * u ; scatter rows to compacted H
    const int ncol  = n0 + ncolw;
    const int mhalf = (lane >> 4) << 3;         // rows 0..7 or 8..15
    #pragma unroll
    for (int i = 0; i < 8; ++i) {
        const int m = m0 + i + mhalf;
        if (m < cnt) {
            const float g = accG[i], u = accU[i];
            const float h = (g / (1.0f + __expf(-g))) * u;
            H[(size_t)(seg + m) * INTER + ncol] = h;
        }
    }
}

// ---------------------------------------------------------------------------
// Stage 2: out[tok] += w * (H_row @ wd[e]). Same pipelined tiling, K=INTER.
// ---------------------------------------------------------------------------
__global__ __launch_bounds__(256) void moe_stage2(
    const float* __restrict__ H,        // [T*K, INTER]
    const float* __restrict__ wd,       // [E, INTER, HIDDEN]
    const int*  __restrict__ offsets,
    const int*  __restrict__ tok_idx,
    const float* __restrict__ tok_w,
    float* __restrict__ out)            // [T, HIDDEN]
{
    const int e   = blockIdx.z;
    const int n0  = blockIdx.y * 128;
    const int seg = offsets[e];
    const int cnt = offsets[e + 1] - seg;
    const int m0  = blockIdx.x * 16;
    if (m0 >= cnt) return;

    __shared__ float Alds[2][16][36];
    __shared__ float Bd[2][32][132];
    __shared__ int   toks[16];
    __shared__ float tw[16];

    const int tid   = threadIdx.x;
    const int lane  = tid & 31;
    const int wave  = tid >> 5;
    const int row16 = lane & 15;
    const int kb    = (lane >> 4) << 1;

    const int a_m = tid >> 3;
    const int a_c = (tid & 7) << 2;
    const int b_c = lane << 2;

    if (tid < 16) {
        const int m = m0 + tid;
        toks[tid] = (m < cnt) ? tok_idx[seg + m] : -1;
        tw[tid]   = (m < cnt) ? tok_w[seg + m]   : 0.0f;
    }
    __syncthreads();

    const float* __restrict__ wdE = wd + (size_t)e * INTER * HIDDEN;

    float4 ra, rd[4];

    auto load_regs = [&](int K0) {
        if (tid < 128) {
            const int mm = m0 + a_m;
            ra = make_float4(0.f, 0.f, 0.f, 0.f);
            if (mm < cnt) ra = *(const float4*)(H + (size_t)(seg + mm) * INTER + K0 + a_c);
        }
        #pragma unroll
        for (int i = 0; i < 4; ++i)
            rd[i] = *(const float4*)(wdE + (size_t)(K0 + wave + 8 * i) * HIDDEN + n0 + b_c);
        if (K0 + 32 < INTER)
            __builtin_prefetch(wdE + (size_t)(K0 + 32 + wave) * HIDDEN + n0 + b_c, 0, 0);
    };
    auto store_regs = [&](int B) {
        if (tid < 128) *(float4*)&Alds[B][a_m][a_c] = ra;
        #pragma unroll
        for (int i = 0; i < 4; ++i)
            *(float4*)&Bd[B][wave + 8 * i][b_c] = rd[i];
    };

    v8f acc = {};
    const int ncolw = (wave << 4) + row16;

    auto compute = [&](int B) {
        #pragma unroll
        for (int kk = 0; kk < 32; kk += 4) {
            v2f a, b;
            a.x = Alds[B][row16][kk + kb];
            a.y = Alds[B][row16][kk + kb + 1];
            b.x = Bd[B][kk + kb][ncolw];
            b.y = Bd[B][kk + kb + 1][ncolw];
            acc = __builtin_amdgcn_wmma_f32_16x16x4_f32(
                false, a, false, b, (short)0, acc, false, false);
        }
    };

    load_regs(0);
    store_regs(0);
    __syncthreads();
    load_regs(32);
    int buf = 0;
    for (int k0 = 0; k0 < INTER; k0 += 32) {
        compute(buf);
        if (k0 + 32 < INTER) {
            store_regs(buf ^ 1);
            if (k0 + 64 < INTER) load_regs(k0 + 64);
        }
        __syncthreads();
        buf ^= 1;
    }

    const int ncol  = n0 + ncolw;
    const int mhalf = (lane >> 4) << 3;
    #pragma unroll
    for (int i = 0; i < 8; ++i) {
        const int m = m0 + i + mhalf;
        if (m < cnt) {
            const int t = toks[i + mhalf];
            const float v = acc[i] * tw[i + mhalf];
            atomicAdd(&out[(size_t)t * HIDDEN + ncol], v);
        }
    }
}

// ---------------------------------------------------------------------------
// Launch
// ---------------------------------------------------------------------------
extern "C" void kernel_launch(void* const* d_in, const int* in_sizes, int n_in,
                              void* d_out, int out_size, void* d_ws, size_t ws_size,
                              hipStream_t stream)
{
    const float* x  = (const float*)d_in[0];   // [2,1024,1024]
    const float* rw = (const float*)d_in[1];   // [8,1024]
    const float* wg = (const float*)d_in[2];   // [8,1024,1408]
    const float* wu = (const float*)d_in[3];   // [8,1024,1408]
    const float* wd = (const float*)d_in[4];   // [8,1408,1024]
    float* out = (float*)d_out;                // [2,1024,1024]

    char* ws = (char*)d_ws;
    int*   counts  = (int*)(ws + 0);            // 8 ints
    int*   cursors = (int*)(ws + 32);           // 8 ints
    int*   offsets = (int*)(ws + 64);           // 9 ints
    int*   topk_i  = (int*)(ws + 128);          // T*2 ints
    float* topk_wv = (float*)(ws + 16512);      // T*2 floats
    int*   tok_i   = (int*)(ws + 32896);        // T*K ints
    float* tok_wv  = (float*)(ws + 49280);      // T*K floats
    float* Hbuf    = (float*)(ws + 131072);     // (T*K) x INTER floats ~ 23.1 MB

    hipMemsetAsync(counts, 0, 32, stream);
    hipMemsetAsync(out, 0, sizeof(float) * TOKENS * HIDDEN, stream);

    moe_router  <<<TOKENS / 8, 256, 0, stream>>>(x, rw, topk_i, topk_wv, counts);
    moe_scan    <<<1, 32, 0, stream>>>(counts, offsets, cursors);
    moe_dispatch<<<TOKENS / 256, 256, 0, stream>>>(topk_i, topk_wv, offsets, cursors,
                                                   tok_i, tok_wv);
    // M tiles sized for worst case (all 2048 tokens on one expert); empty tiles exit.
    moe_stage1<<<dim3(TOKENS / 16, INTER / 128, NEXPERTS), 256, 0, stream>>>(
        x, wg, wu, offsets, tok_i, Hbuf);
    moe_stage2<<<dim3(TOKENS / 16, HIDDEN / 128, NEXPERTS), 256, 0, stream>>>(
        Hbuf, wd, offsets, tok_i, tok_wv, out);
}